// Citeseer_GCN_14937896255790
// MI455X (gfx1250) — compile-verified
//
#include <hip/hip_runtime.h>
#include <hip/hip_bf16.h>

#define NN      50000
#define EE      800000
#define IN_DIM  512
#define HID     64
#define OUT_DIM 40

// Padded row stride (in float2) for the K-paired W1 LDS image.
// 80 float2 = 160 dwords == 32 (mod 64) -> lanes 16-31 land on banks 32..63,
// so a wave's ds_load_b64 covers all 64 LDS banks exactly once.
#define PKS 80

typedef __attribute__((ext_vector_type(2))) float v2f;
typedef __attribute__((ext_vector_type(8))) float v8f;

// ---------------- K0: deg = 1.0 (self loop) ----------------
__global__ void k_init_deg(float* __restrict__ deg) {
    int i = blockIdx.x * 256 + threadIdx.x;
    if (i < NN) deg[i] = 1.0f;
}

// ---------------- K1: deg[dst] += 1 ----------------
__global__ void k_degree(const long long* __restrict__ edges, float* __restrict__ deg) {
    int e = blockIdx.x * 256 + threadIdx.x;
    if (e < EE) {
        long long d = edges[EE + e];
        unsafeAtomicAdd(&deg[d], 1.0f);
    }
}

// ---------------- K2: dinv = rsqrt(deg) in place ----------------
__global__ void k_dinv(float* __restrict__ deg) {
    int i = blockIdx.x * 256 + threadIdx.x;
    if (i < NN) {
        float d = deg[i];
        deg[i] = (d > 0.0f) ? rsqrtf(d) : 0.0f;
    }
}

// ---------------- K3: h = X @ W1 (WMMA f32 16x16x4), agg = dinv^2 * h -------
// Block = 256 threads = 8 waves; each wave owns 16 rows x all 64 cols.
// W1 staged into LDS in K-paired float2 layout:
//   sW2[kp * PKS + n] = { W1[2kp][n], W1[2kp+1][n] }   (kp = 0..255, n = 0..63)
// so each lane's B fragment is a single aligned ds_load_b64 (no shuffle movs).
__global__ __launch_bounds__(256) void k_gemm1(
    const float* __restrict__ X, const float* __restrict__ W1,
    const float* __restrict__ dinv,
    float* __restrict__ h, float* __restrict__ agg) {
    extern __shared__ v2f sW2[];   // (IN_DIM/2) * PKS float2 = 160 KB

    const int tid = threadIdx.x;

    // Stage + swizzle W1 -> LDS. 16384 pairs / 256 threads = 64 per thread.
    // Consecutive tids hit consecutive n -> coalesced 256B global reads.
    for (int idx = tid; idx < (IN_DIM / 2) * HID; idx += 256) {
        int kp = idx >> 6;        // K pair index
        int n  = idx & 63;        // column
        v2f w;
        w.x = W1[(size_t)(2 * kp + 0) * HID + n];
        w.y = W1[(size_t)(2 * kp + 1) * HID + n];
        sW2[(size_t)kp * PKS + n] = w;
    }
    __syncthreads();

    const int wave  = tid >> 5;
    const int lane  = tid & 31;
    const int lrow  = lane & 15;        // M (for A) / N (for B) index 0..15
    const int kh    = lane >> 4;        // 0: K-pair (k0,k0+1); 1: (k0+2,k0+3)
    const int r0    = blockIdx.x * 128 + wave * 16;

    // Clamp row for loads so EXEC stays all-1s for WMMA; guard only stores.
    int row  = r0 + lrow;
    int rowc = (row < NN) ? row : (NN - 1);
    const float* Xr = X + (size_t)rowc * IN_DIM;

    // Loop-invariant part of the B address: kp = k0/2 + kh, col = lrow + 16t.
    const v2f* bp = sW2 + (size_t)kh * PKS + lrow;

    v8f acc0 = {}, acc1 = {}, acc2 = {}, acc3 = {};

    for (int k0 = 0; k0 < IN_DIM; k0 += 4) {
        const int kb = k0 + kh * 2;               // even -> 8B-aligned pair
        v2f a = *(const v2f*)(Xr + kb);           // global_load_b64

        const v2f* bk = bp + (size_t)(k0 >> 1) * PKS;
        v2f b0 = bk[0];                           // ds_load_b64, cols  0..15
        v2f b1 = bk[16];                          //               16..31
        v2f b2 = bk[32];                          //               32..47
        v2f b3 = bk[48];                          //               48..63

        acc0 = __builtin_amdgcn_wmma_f32_16x16x4_f32(false, a, false, b0, (short)0, acc0, false, false);
        acc1 = __builtin_amdgcn_wmma_f32_16x16x4_f32(false, a, false, b1, (short)0, acc1, false, false);
        acc2 = __builtin_amdgcn_wmma_f32_16x16x4_f32(false, a, false, b2, (short)0, acc2, false, false);
        acc3 = __builtin_amdgcn_wmma_f32_16x16x4_f32(false, a, false, b3, (short)0, acc3, false, false);
    }

    // D layout (16x16 f32): VGPR j -> M = j (lanes 0-15) or j+8 (lanes 16-31),
    // N = lane & 15.
    const int rsel = (lane < 16) ? 0 : 8;
    #pragma unroll
    for (int j = 0; j < 8; ++j) {
        int rr = r0 + rsel + j;
        if (rr < NN) {
            float dv = dinv[rr];
            float c2 = dv * dv;           // self-loop coefficient 1/deg
            size_t base = (size_t)rr * HID + lrow;
            h[base +  0] = acc0[j];  agg[base +  0] = c2 * acc0[j];
            h[base + 16] = acc1[j];  agg[base + 16] = c2 * acc1[j];
            h[base + 32] = acc2[j];  agg[base + 32] = c2 * acc2[j];
            h[base + 48] = acc3[j];  agg[base + 48] = c2 * acc3[j];
        }
    }
}

// ---------------- K4: edge scatter-add, one wave per edge ----------------
__global__ __launch_bounds__(256) void k_aggregate(
    const long long* __restrict__ edges, const float* __restrict__ dinv,
    const float* __restrict__ h, float* __restrict__ agg) {
    int tid  = threadIdx.x;
    int e    = blockIdx.x * 8 + (tid >> 5);
    int lane = tid & 31;
    if (e < EE) {
        long long s = edges[e];
        long long d = edges[EE + e];
        float c = dinv[s] * dinv[d];
        const float* hs = h + (size_t)s * HID;
        float* ad = agg + (size_t)d * HID;
        unsafeAtomicAdd(&ad[lane +  0], c * hs[lane +  0]);
        unsafeAtomicAdd(&ad[lane + 32], c * hs[lane + 32]);
    }
}

// ---------------- K5: out = relu(agg + b1) @ Wfc^T + bfc ----------------
__global__ __launch_bounds__(256) void k_fc(
    const float* __restrict__ agg, const float* __restrict__ b1,
    const float* __restrict__ Wfc, const float* __restrict__ bfc,
    float* __restrict__ out) {
    __shared__ float sW[OUT_DIM * HID];
    __shared__ float sb1[HID];
    __shared__ float sbf[OUT_DIM];
    int tid = threadIdx.x;
    for (int i = tid; i < OUT_DIM * HID; i += 256) sW[i] = Wfc[i];
    if (tid < HID) sb1[tid] = b1[tid];
    if (tid < OUT_DIM) sbf[tid] = bfc[tid];
    __syncthreads();

    int n = blockIdx.x * 256 + tid;
    if (n >= NN) return;

    const float* ar = agg + (size_t)n * HID;
    float rowv[HID];
    #pragma unroll
    for (int k = 0; k < HID; ++k) {
        float v = ar[k] + sb1[k];
        rowv[k] = (v > 0.0f) ? v : 0.0f;
    }

    float* orow = out + (size_t)n * OUT_DIM;
    for (int o = 0; o < OUT_DIM; ++o) {
        float acc = sbf[o];
        #pragma unroll
        for (int k = 0; k < HID; ++k) acc += rowv[k] * sW[o * HID + k];
        orow[o] = acc;
    }
}

// ---------------- host launch ----------------
extern "C" void kernel_launch(void* const* d_in, const int* in_sizes, int n_in,
                              void* d_out, int out_size, void* d_ws, size_t ws_size,
                              hipStream_t stream) {
    const float*     X     = (const float*)d_in[0];
    const long long* edges = (const long long*)d_in[1];
    const float*     W1    = (const float*)d_in[2];
    const float*     b1    = (const float*)d_in[3];
    const float*     Wfc   = (const float*)d_in[4];
    const float*     bfc   = (const float*)d_in[5];
    float*           out   = (float*)d_out;

    // Workspace layout (floats): deg/dinv [NN pad], h [NN*HID], agg [NN*HID]
    const size_t NPAD = 50176;  // NN rounded up to multiple of 256
    float* deg = (float*)d_ws;
    float* h   = deg + NPAD;
    float* agg = h + (size_t)NN * HID;

    const size_t ldsBytes = (size_t)(IN_DIM / 2) * PKS * sizeof(float) * 2; // 160 KB

    k_init_deg<<<(NN + 255) / 256, 256, 0, stream>>>(deg);
    k_degree<<<(EE + 255) / 256, 256, 0, stream>>>(edges, deg);
    k_dinv<<<(NN + 255) / 256, 256, 0, stream>>>(deg);
    k_gemm1<<<(NN + 127) / 128, 256, ldsBytes, stream>>>(X, W1, deg, h, agg);
    k_aggregate<<<(EE + 7) / 8, 256, 0, stream>>>(edges, deg, h, agg);
    k_fc<<<(NN + 255) / 256, 256, 0, stream>>>(agg, b1, Wfc, bfc, out);
}